// word_choser_23974507446846
// MI455X (gfx1250) — compile-verified
//
#include <hip/hip_runtime.h>
#include <math.h>

#define HDIM 1024
#define EDIM 1024
#define TDIM 4096
#define VDIM 50257
#define INPD 2049   // E + H + 1

typedef __attribute__((ext_vector_type(2))) float v2f;
typedef __attribute__((ext_vector_type(8))) float v8f;

// ---------------------------------------------------------------------------
// Kernel 1a: initialize x = [sen_emb | zeros (colsum accumulator) | pos]
// ---------------------------------------------------------------------------
__global__ void init_x_kernel(const float* __restrict__ sen_emb,
                              const int* __restrict__ pos,
                              float* __restrict__ x) {
    int i = blockIdx.x * blockDim.x + threadIdx.x;
    if (i < EDIM) {
        x[i] = sen_emb[i];
    } else if (i < EDIM + HDIM) {
        x[i] = 0.0f;               // colsum accumulator, filled by atomics
    } else if (i == EDIM + HDIM) {
        x[i] = (float)pos[0];
    }
}

// ---------------------------------------------------------------------------
// Kernel 1b: x[1024+e] += column sums of hiddens (split over T for parallelism)
// grid = (EDIM/256, 32), block = 256
// ---------------------------------------------------------------------------
__global__ void colsum_kernel(const float* __restrict__ hiddens,
                              float* __restrict__ x) {
    int e  = blockIdx.x * 256 + threadIdx.x;
    int t0 = blockIdx.y * (TDIM / 32);
    float s = 0.0f;
    for (int t = t0; t < t0 + (TDIM / 32); ++t)
        s += hiddens[(size_t)t * EDIM + e];
    atomicAdd(&x[EDIM + e], s);    // global_atomic_add_f32
}

// ---------------------------------------------------------------------------
// Kernel 2: LSTM gates. One wave32 per output row; coalesced row reads,
// __shfl_xor wave reduction. Writes h[1024].
// ---------------------------------------------------------------------------
__global__ void gates_kernel(const float* __restrict__ x,
                             const float* __restrict__ cur_h,
                             const float* __restrict__ cur_cell,
                             const float* Wi0, const float* Wh0, const float* bi0, const float* bh0,
                             const float* Wi1, const float* Wh1, const float* bi1, const float* bh1,
                             const float* Wi2, const float* Wh2, const float* bi2, const float* bh2,
                             const float* Wi3, const float* Wh3, const float* bi3, const float* bh3,
                             float* __restrict__ h_out) {
    int wave = (blockIdx.x * blockDim.x + threadIdx.x) >> 5;
    int lane = threadIdx.x & 31;
    if (wave >= HDIM) return;      // wave-uniform

    const float* Wi[4] = {Wi0, Wi1, Wi2, Wi3};
    const float* Wh[4] = {Wh0, Wh1, Wh2, Wh3};
    const float* bi[4] = {bi0, bi1, bi2, bi3};
    const float* bh[4] = {bh0, bh1, bh2, bh3};

    float gv[4];
#pragma unroll
    for (int g = 0; g < 4; ++g) {
        float acc = 0.0f;
        const float* wrow = Wi[g] + (size_t)wave * INPD;
        for (int k = lane; k < INPD; k += 32) acc += wrow[k] * x[k];
        const float* hrow = Wh[g] + (size_t)wave * HDIM;
        for (int k = lane; k < HDIM; k += 32) acc += hrow[k] * cur_h[k];
        for (int off = 16; off > 0; off >>= 1)
            acc += __shfl_xor(acc, off, 32);
        gv[g] = acc + bi[g][wave] + bh[g][wave];
    }

    if (lane == 0) {
        float iv = 1.0f / (1.0f + expf(-gv[0]));
        float fv = 1.0f / (1.0f + expf(-gv[1]));
        float gg = tanhf(gv[2]);
        float ov = 1.0f / (1.0f + expf(-gv[3]));
        float c  = fv * cur_cell[wave] + iv * gg;
        h_out[wave] = ov * tanhf(c);
    }
}

// ---------------------------------------------------------------------------
// Kernel 3: logits = h @ dim_out via V_WMMA_F32_16X16X4_F32.
// One wave -> TWO 16-column tiles (32 columns) with independent accumulators.
// A fragment (shared by both tiles): M=0 row carries h[k..k+3] (lane 0 holds
// K=0/1, lane 16 holds K=2/3 per the 16x4 f32 layout), all other rows zero.
// B fragments: VGPR0 = rows K0 (lanes 0-15) / K2 (lanes 16-31), VGPR1 = K1/K3
// -> 16-lane-contiguous dword loads of dim_out. Out-of-range columns are
// CLAMPED (not exec-masked): their D columns are garbage but never stored,
// so the hot loop has zero divergence / exec manipulation.
// D: row M=0 lands in acc[0], lanes 0-15 = tile columns.
// ---------------------------------------------------------------------------
__global__ void logits_wmma_kernel(const float* __restrict__ h,
                                   const float* __restrict__ dim_out,
                                   float* __restrict__ logits) {
    __shared__ float sh[HDIM];
    for (int i = threadIdx.x; i < HDIM; i += blockDim.x) sh[i] = h[i];
    __syncthreads();

    int wave = (blockIdx.x * blockDim.x + threadIdx.x) >> 5;
    int lane = threadIdx.x & 31;
    int v0   = wave * 32;
    if (v0 >= VDIM) return;        // wave-uniform

    int nsub  = lane & 15;                 // column within tile
    int khalf = (lane >> 4) * 2;           // 0 for lanes 0-15, 2 for 16-31
    int col0  = v0 + nsub;                 // tile 0 column
    int col1  = v0 + 16 + nsub;            // tile 1 column
    int c0    = (col0 < VDIM) ? col0 : (VDIM - 1);   // clamp: always in-bounds
    int c1    = (col1 < VDIM) ? col1 : (VDIM - 1);
    // only lanes 0 and 16 carry the M=0 row of A
    float amask = (nsub == 0) ? 1.0f : 0.0f;

    const float* __restrict__ p0 = dim_out + c0;
    const float* __restrict__ p1 = dim_out + c1;

    v8f acc0 = {};
    v8f acc1 = {};
#pragma unroll 4
    for (int kk = 0; kk < HDIM; kk += 4) {
        v2f a, b0, b1;
        a.x = sh[kk + khalf]     * amask;
        a.y = sh[kk + khalf + 1] * amask;
        size_t base = (size_t)(kk + khalf) * VDIM;
        b0.x = p0[base];
        b0.y = p0[base + VDIM];
        b1.x = p1[base];
        b1.y = p1[base + VDIM];
        acc0 = __builtin_amdgcn_wmma_f32_16x16x4_f32(
            false, a, false, b0, (short)0, acc0, false, false);
        acc1 = __builtin_amdgcn_wmma_f32_16x16x4_f32(
            false, a, false, b1, (short)0, acc1, false, false);
    }

    if (lane < 16) {
        if (col0 < VDIM) logits[col0] = acc0[0];
        if (col1 < VDIM) logits[col1] = acc1[0];
    }
}

// ---------------------------------------------------------------------------
// Kernel 4: softmax reduction (max + sum of exp), single block
// ---------------------------------------------------------------------------
__global__ void softmax_reduce_kernel(const float* __restrict__ logits,
                                      float* __restrict__ red) {
    __shared__ float sm[256];
    int tid = threadIdx.x;
    float m = -3.402823466e38f;
    for (int i = tid; i < VDIM; i += 256) m = fmaxf(m, logits[i]);
    sm[tid] = m; __syncthreads();
    for (int s = 128; s > 0; s >>= 1) {
        if (tid < s) sm[tid] = fmaxf(sm[tid], sm[tid + s]);
        __syncthreads();
    }
    float gmax = sm[0]; __syncthreads();
    float acc = 0.0f;
    for (int i = tid; i < VDIM; i += 256) acc += expf(logits[i] - gmax);
    sm[tid] = acc; __syncthreads();
    for (int s = 128; s > 0; s >>= 1) {
        if (tid < s) sm[tid] += sm[tid + s];
        __syncthreads();
    }
    if (tid == 0) { red[0] = gmax; red[1] = sm[0]; }
}

// ---------------------------------------------------------------------------
// Kernel 5: softmax normalize
// ---------------------------------------------------------------------------
__global__ void softmax_norm_kernel(const float* __restrict__ logits,
                                    const float* __restrict__ red,
                                    float* __restrict__ out) {
    int v = blockIdx.x * blockDim.x + threadIdx.x;
    if (v < VDIM) out[v] = expf(logits[v] - red[0]) / red[1];
}

// ---------------------------------------------------------------------------
extern "C" void kernel_launch(void* const* d_in, const int* in_sizes, int n_in,
                              void* d_out, int out_size, void* d_ws, size_t ws_size,
                              hipStream_t stream) {
    const float* sen_emb  = (const float*)d_in[0];
    const float* hiddens  = (const float*)d_in[1];
    const float* dim_out  = (const float*)d_in[2];
    const float* Wi[4]    = {(const float*)d_in[3],  (const float*)d_in[7],
                             (const float*)d_in[11], (const float*)d_in[15]};
    const float* Wh[4]    = {(const float*)d_in[4],  (const float*)d_in[8],
                             (const float*)d_in[12], (const float*)d_in[16]};
    const float* bi[4]    = {(const float*)d_in[5],  (const float*)d_in[9],
                             (const float*)d_in[13], (const float*)d_in[17]};
    const float* bh[4]    = {(const float*)d_in[6],  (const float*)d_in[10],
                             (const float*)d_in[14], (const float*)d_in[18]};
    const float* cur_h    = (const float*)d_in[19];
    const float* cur_cell = (const float*)d_in[20];
    const int*   pos      = (const int*)d_in[21];

    float* ws      = (float*)d_ws;
    float* x       = ws;                 // 2049 floats (pad to 2064)
    float* hvec    = ws + 2064;          // 1024
    float* logits  = ws + 2064 + 1024;   // 50257 (pad to 50272)
    float* red     = ws + 2064 + 1024 + 50272;  // 2

    // 1) build x = [sen_emb | colsum(hiddens) | pos]
    init_x_kernel<<<(INPD + 255) / 256, 256, 0, stream>>>(sen_emb, pos, x);
    dim3 cs_grid(EDIM / 256, 32);
    colsum_kernel<<<cs_grid, 256, 0, stream>>>(hiddens, x);

    // 2) LSTM gates -> h   (1024 waves = 128 blocks of 8 waves)
    gates_kernel<<<128, 256, 0, stream>>>(
        x, cur_h, cur_cell,
        Wi[0], Wh[0], bi[0], bh[0],
        Wi[1], Wh[1], bi[1], bh[1],
        Wi[2], Wh[2], bi[2], bh[2],
        Wi[3], Wh[3], bi[3], bh[3],
        hvec);

    // 3) logits = h @ dim_out via WMMA f32 16x16x4, 32 columns per wave
    int waves  = (VDIM + 31) / 32;         // 1571
    int blocks = (waves + 7) / 8;          // 197 blocks of 8 waves
    logits_wmma_kernel<<<blocks, 256, 0, stream>>>(hvec, dim_out, logits);

    // 4) softmax
    softmax_reduce_kernel<<<1, 256, 0, stream>>>(logits, red);
    softmax_norm_kernel<<<(VDIM + 255) / 256, 256, 0, stream>>>(
        logits, red, (float*)d_out);
}